// CTCLossEndpointCpu_53635551593260
// MI455X (gfx1250) — compile-verified
//
#include <hip/hip_runtime.h>
#include <stdint.h>

#define NEG_INF  (-1e30f)
#define LOG2E_F  1.4426950408889634f
#define LN2_F    0.6931471805599453f

#define CTC_V    256      // vocab size (fixed by problem)
#define CTC_L    256      // max label length supported by this build
#define CTC_SMAX (2 * CTC_L + 1)          // 513 extended states
#define ALPHA_W  (2 + CTC_SMAX + 1)       // 516: 2 guard cells + pad (8B align of row 1)
#define PF_D     16       // async prefetch ring depth (timesteps resident in LDS)

// ---------------------------------------------------------------------------
// Kernel 1: lse2[b*T + t] = log2( sum_v 2^(logits[t,b,v]*log2e) )
// One wave32 per (t,b) row; float4 loads; shuffle reductions. Memory-bound.
// Only consumed as a *sum over t* in the epilogue of kernel 2 (the per-step
// normalization is factored out of the recursion by shift-equivariance).
// ---------------------------------------------------------------------------
__global__ __launch_bounds__(256) void ctc_lse2_kernel(
    const float* __restrict__ logits,   // [T,B,V]
    float* __restrict__ lse2,           // [B,T]
    int T, int B)
{
  const int lane = threadIdx.x & 31;
  const int row  = (int)((blockIdx.x * blockDim.x + threadIdx.x) >> 5); // t*B + b
  if (row >= T * B) return;

  const float4* r4 = (const float4*)(logits + (size_t)row * CTC_V);
  float4 a = r4[lane];
  float4 c = r4[lane + 32];

  float m = fmaxf(fmaxf(fmaxf(a.x, a.y), fmaxf(a.z, a.w)),
                  fmaxf(fmaxf(c.x, c.y), fmaxf(c.z, c.w)));
#pragma unroll
  for (int off = 16; off > 0; off >>= 1)
    m = fmaxf(m, __shfl_xor(m, off, 32));

  float s =
      exp2f((a.x - m) * LOG2E_F) + exp2f((a.y - m) * LOG2E_F) +
      exp2f((a.z - m) * LOG2E_F) + exp2f((a.w - m) * LOG2E_F) +
      exp2f((c.x - m) * LOG2E_F) + exp2f((c.y - m) * LOG2E_F) +
      exp2f((c.z - m) * LOG2E_F) + exp2f((c.w - m) * LOG2E_F);
#pragma unroll
  for (int off = 16; off > 0; off >>= 1)
    s += __shfl_xor(s, off, 32);

  if (lane == 0) {
    int t = row / B, b = row - t * B;
    lse2[(size_t)b * T + t] = fmaf(m, LOG2E_F, log2f(s));
  }
}

// ---------------------------------------------------------------------------
// Kernel 2: CTC forward recursion on UNNORMALIZED base-2 log-probs.
// One block per batch element. Thread tid owns states (2tid, 2tid+1); their
// union stencil alpha[1+2tid .. 3+2tid] is fetched as ds_load_b64+ds_load_b32
// and the two results written as one ds_store_b64. Trailing blank s=2L is
// handled by a lane of wave 1. Logits rows stream through a 16-deep LDS ring
// filled with CDNA5 global_load_async_to_lds_b128 (ASYNCcnt) from wave 0:
// 15 rows (~2.3 us) of lead fully hides idle-machine HBM latency behind the
// ~150-cycle steps; slot t is provably complete at constant s_wait_asynccnt 28.
// ---------------------------------------------------------------------------
__global__ __launch_bounds__(256) void ctc_alpha_kernel(
    const float* __restrict__ logits,    // [T,B,V]
    const int*   __restrict__ labels,    // [B,L]
    const int*   __restrict__ label_len, // [B]
    const int*   __restrict__ logit_len, // [B]
    const float* __restrict__ lse2,      // [B,T]
    float*       __restrict__ loss,      // [B]
    int T, int B, int L)
{
  __shared__ float rowbuf[PF_D * CTC_V];   // 16 KiB ring of logits rows
  __shared__ float alpha[2][ALPHA_W];      // ping-pong + guards [0],[1]
  __shared__ float red[256];               // epilogue reduction scratch

  const int b    = blockIdx.x;
  const int tid  = threadIdx.x;
  const int lane = tid & 31;
  const int S    = 2 * L + 1;

  // init both alpha buffers (incl. guard cells) to log-zero
  for (int i = tid; i < 2 * ALPHA_W; i += 256)
    (&alpha[0][0])[i] = NEG_INF;

  // per-thread label info: odd state s=2tid+1 emits labels[b][tid]
  int lab = 0, skipf = 0;
  const bool has_pair = (tid < L);         // states 2tid,2tid+1 valid
  if (has_pair) {
    lab = labels[(size_t)b * L + tid] & (CTC_V - 1);
    int prv = (tid > 0) ? (labels[(size_t)b * L + tid - 1] & (CTC_V - 1)) : -1;
    skipf = (lab != prv);                  // s==1 skip reads NEG_INF guard: harmless
  }
  const bool has_tail = (tid == 32) && (2 * L == S - 1);  // trailing blank s=2L

  int Tend = logit_len[b];
  if (Tend > T) Tend = T;
  if (Tend < 1) Tend = 1;

  const unsigned lds_base = (unsigned)(uintptr_t)(void*)&rowbuf[0];
  const unsigned voff     = (unsigned)lane * 16u;  // 32 lanes x 16B = 512B/issue

  const size_t rowStride = (size_t)B * CTC_V;      // floats between rows t,t+1

  // prologue: wave 0 preloads PF_D-1 rows (2 async b128 issues per row = 1 KiB)
  if (tid < 32) {
    for (int j = 0; j < PF_D - 1; ++j) {
      int r = (j < Tend) ? j : (Tend - 1);
      const float* g = logits + (size_t)r * rowStride + (size_t)b * CTC_V;
      unsigned la = lds_base + (unsigned)j * (CTC_V * 4u) + voff;
      asm volatile(
          "global_load_async_to_lds_b128 %0, %1, %2\n\t"
          "global_load_async_to_lds_b128 %0, %1, %2 offset:512"
          :: "v"(la), "v"(voff), "s"(g) : "memory");
    }
  }

  // running pointer for the in-loop refill row: r = min(t+PF_D-1, Tend-1)
  int r0 = (PF_D - 1 < Tend) ? (PF_D - 1) : (Tend - 1);
  const float* gnext = logits + (size_t)r0 * rowStride + (size_t)b * CTC_V;

  // per-iteration pipeline helper: (a) wave0 proves row t landed
  // (outstanding <= 2*(PF_D-2) = 28), (b) barrier makes it visible to all
  // waves AND frees the slot consumed last step, (c) wave0 refills that slot.
  auto pipeline = [&](int t) {
    if (tid < 32) {
      asm volatile("s_wait_asynccnt 28" ::: "memory");
    }
    __syncthreads();
    if (tid < 32) {
      unsigned la = lds_base +
                    (unsigned)((t + PF_D - 1) & (PF_D - 1)) * (CTC_V * 4u) + voff;
      asm volatile(
          "global_load_async_to_lds_b128 %0, %1, %2\n\t"
          "global_load_async_to_lds_b128 %0, %1, %2 offset:512"
          :: "v"(la), "v"(voff), "s"(gnext) : "memory");
    }
    if (t + PF_D < Tend) gnext += rowStride;
  };

  // -------- t = 0 (peeled): alpha0 init ---------------------------------
  pipeline(0);
  {
    const float* row = &rowbuf[0];
    const float eb = row[0]   * LOG2E_F;
    const float eo = row[lab] * LOG2E_F;
    if (has_pair) {
      float v0 = (tid == 0) ? eb : NEG_INF;   // s=0
      float v1 = (tid == 0) ? eo : NEG_INF;   // s=1
      *(float2*)&alpha[0][2 + 2 * tid] = make_float2(v0, v1);
    }
    if (has_tail) alpha[0][2 + 2 * L] = NEG_INF;
  }

  // -------- steady-state recursion --------------------------------------
  for (int t = 1; t < Tend; ++t) {
    pipeline(t);

    const float* row = &rowbuf[(t & (PF_D - 1)) * CTC_V];
    const int    wb  = t & 1;       // write buffer
    const int    rb  = wb ^ 1;      // read buffer

    const float eb = row[0]   * LOG2E_F;   // blank emission (LDS broadcast)
    const float eo = row[lab] * LOG2E_F;   // this thread's label emission

    if (has_pair) {
      // stencil union: x1 = alpha[1+2tid], {x2,x3} = alpha[2+2tid..3+2tid]
      float  x1  = alpha[rb][1 + 2 * tid];
      float2 x23 = *(const float2*)&alpha[rb][2 + 2 * tid];
      // even state s=2tid: blank, paths (s, s-1)
      float m0 = fmaxf(x23.x, x1);
      float v0 = m0 + log2f(exp2f(x23.x - m0) + exp2f(x1 - m0)) + eb;
      // odd state s=2tid+1: paths (s, s-1, skip? s-2)
      float a2 = skipf ? x1 : NEG_INF;
      float m1 = fmaxf(x23.y, fmaxf(x23.x, a2));
      float v1 = m1 + log2f(exp2f(x23.y - m1) + exp2f(x23.x - m1) +
                            exp2f(a2 - m1)) + eo;
      *(float2*)&alpha[wb][2 + 2 * tid] = make_float2(v0, v1);
    }
    if (has_tail) {
      float a0 = alpha[rb][2 + 2 * L];       // s=2L
      float a1 = alpha[rb][1 + 2 * L];       // s=2L-1
      float m  = fmaxf(a0, a1);
      alpha[wb][2 + 2 * L] =
          m + log2f(exp2f(a0 - m) + exp2f(a1 - m)) + eb;
    }
  }

  // epilogue: correction sum C = sum_{t<Tend} lse2[b,t] (factored normalizer)
  float part = 0.0f;
  for (int i = tid; i < Tend; i += 256)
    part += lse2[(size_t)b * T + i];
  red[tid] = part;
  __syncthreads();   // also publishes the final alpha writes

  if (tid == 0) {
    double csum = 0.0;
    for (int i = 0; i < 256; ++i) csum += (double)red[i];

    int fin = (Tend - 1) & 1;
    int len = label_len[b];
    if (len < 0) len = 0;
    if (len > L) len = L;
    int end = 2 * len;
    int em1 = (end > 0) ? end - 1 : 0;
    float x = alpha[fin][2 + end];
    float y = alpha[fin][2 + em1];
    float m = fmaxf(x, y);
    float ll2_hat = m + log2f(exp2f(x - m) + exp2f(y - m));  // unnormalized
    loss[b] = (float)((csum - (double)ll2_hat) * (double)LN2_F);
  }
}

// ---------------------------------------------------------------------------
// Kernel 3: deterministic fixed-order mean of per-sample losses.
// ---------------------------------------------------------------------------
__global__ void ctc_mean_kernel(const float* __restrict__ loss,
                                float* __restrict__ out, int B)
{
  if (threadIdx.x == 0 && blockIdx.x == 0) {
    float acc = 0.0f;
    for (int i = 0; i < B; ++i) acc += loss[i];
    out[0] = acc / (float)B;
  }
}

// ---------------------------------------------------------------------------
extern "C" void kernel_launch(void* const* d_in, const int* in_sizes, int n_in,
                              void* d_out, int out_size, void* d_ws, size_t ws_size,
                              hipStream_t stream) {
  const int*   labels    = (const int*)d_in[0];    // [B,L]
  const float* logits    = (const float*)d_in[1];  // [T,B,V]
  const int*   label_len = (const int*)d_in[2];    // [B]
  const int*   logit_len = (const int*)d_in[3];    // [B]

  const int B = in_sizes[2];
  const int L = in_sizes[0] / B;
  const int T = in_sizes[1] / (B * CTC_V);

  float* ws_loss = (float*)d_ws;                   // B floats
  float* ws_lse  = (float*)((char*)d_ws + 256);    // B*T floats (lse2, [B,T])

  const int rows    = T * B;
  const int blocks1 = (rows + 7) / 8;              // 8 wave-rows per 256-thr block
  ctc_lse2_kernel<<<blocks1, 256, 0, stream>>>(logits, ws_lse, T, B);
  ctc_alpha_kernel<<<B, 256, 0, stream>>>(logits, labels, label_len, logit_len,
                                          ws_lse, ws_loss, T, B, L);
  ctc_mean_kernel<<<1, 32, 0, stream>>>(ws_loss, (float*)d_out, B);

  (void)n_in; (void)out_size; (void)ws_size;
}